// S1RecurrentAttentionCell_72739566125770
// MI455X (gfx1250) — compile-verified
//
#include <hip/hip_runtime.h>

typedef __attribute__((ext_vector_type(2))) float v2f;
typedef __attribute__((ext_vector_type(8))) float v8f;

#define DK 128
#define DV 128

// One workgroup (8 wave32) per (b,h) pair. Wave w owns state columns
// [16w, 16w+16). Single pass over the 128x128 fp32 state:
//   a1 = k^T S0, a2 = q^T S0   (via v_wmma_f32_16x16x4_f32, A row0=q row1=k)
//   delta = (v - d*a1)*beta ;  out = d*a2 + (q.k)*delta ; S' = d*S0 + k (x) delta
__global__ __launch_bounds__(256) void gdr_cell_kernel(
    const float* __restrict__ q, const float* __restrict__ k,
    const float* __restrict__ v, const float* __restrict__ g,
    const float* __restrict__ beta, const float* __restrict__ S0,
    float* __restrict__ out, int BH)
{
    __shared__ float sh[3 * DK];   // [0,128): q  [128,256): k  [256,384): v

    const int bh   = blockIdx.x;
    const int t    = threadIdx.x;
    const int lane = t & 31;
    const int wave = t >> 5;
    const int m    = lane & 15;   // column within wave tile / A-row id
    const int half = lane >> 4;   // selects K pair (0,1) vs (2,3)
    const int col  = wave * 16 + m;

    const float* qp = q + (size_t)bh * DK;
    const float* kp = k + (size_t)bh * DK;
    const float* vp = v + (size_t)bh * DV;
    if (t < DK) { sh[t] = qp[t]; sh[DK + t] = kp[t]; sh[2 * DK + t] = vp[t]; }
    __syncthreads();

    const float decay = __expf(g[bh]);
    const float bta   = beta[bh];

    // A-operand source: row 0 = q, row 1 = k, rows 2..15 = 0.
    // Unconditional LDS address (selected once), value zeroed via cndmask.
    const bool isA   = (m < 2);
    const int  abase = (m == 1) ? DK : 0;

    const float* __restrict__ Sb = S0 + (size_t)bh * (DK * DV);

    // --- pass: load state once (kept in VGPRs), accumulate both matvecs via WMMA
    float st[64];
    v8f acc = {};
    #pragma unroll
    for (int kc = 0; kc < 32; ++kc) {
        const int r0 = 4 * kc + 2 * half;
        const float s0 = __builtin_nontemporal_load(&Sb[r0 * DV + col]);       // B vgpr0: K=2*half
        const float s1 = __builtin_nontemporal_load(&Sb[(r0 + 1) * DV + col]); // B vgpr1: K=2*half+1
        st[2 * kc]     = s0;
        st[2 * kc + 1] = s1;
        const float a0raw = sh[abase + r0];       // consecutive -> ds_load_b64
        const float a1raw = sh[abase + r0 + 1];
        const float a0 = isA ? a0raw : 0.0f;      // v_cndmask, no exec branching
        const float a1 = isA ? a1raw : 0.0f;
        v2f A; A[0] = a0; A[1] = a1;
        v2f B; B[0] = s0; B[1] = s1;
        acc = __builtin_amdgcn_wmma_f32_16x16x4_f32(
                  false, A, false, B, (short)0, acc, false, false);
    }

    // --- q.k : per-lane partial over 4 elements + butterfly reduce
    float pq = sh[lane * 4 + 0] * sh[DK + lane * 4 + 0]
             + sh[lane * 4 + 1] * sh[DK + lane * 4 + 1]
             + sh[lane * 4 + 2] * sh[DK + lane * 4 + 2]
             + sh[lane * 4 + 3] * sh[DK + lane * 4 + 3];
    #pragma unroll
    for (int off = 16; off > 0; off >>= 1) pq += __shfl_xor(pq, off, 32);
    const float qk = pq;

    // C layout: lanes 0..15 hold M=0 (a2) in acc[0], M=1 (a1) in acc[1], N=lane.
    const float a2    = acc[0];
    const float a1s   = acc[1];
    const float kvmem = decay * a1s;
    const float delta = (sh[2 * DK + col] - kvmem) * bta;  // valid on lanes 0..15
    const float o     = decay * a2 + qk * delta;
    const float delta_b = __shfl(delta, m, 32);            // broadcast to upper half

    if (half == 0) out[(size_t)bh * DV + col] = o;

    // --- rank-1 update + single streamed (nontemporal) store of S'
    float* __restrict__ So = out + (size_t)BH * DV + (size_t)bh * (DK * DV);
    #pragma unroll
    for (int kc = 0; kc < 32; ++kc) {
        const int r0 = 4 * kc + 2 * half;
        const float ns0 = decay * st[2 * kc]     + sh[DK + r0]     * delta_b;
        const float ns1 = decay * st[2 * kc + 1] + sh[DK + r0 + 1] * delta_b;
        __builtin_nontemporal_store(ns0, &So[r0 * DV + col]);
        __builtin_nontemporal_store(ns1, &So[(r0 + 1) * DV + col]);
    }
}

extern "C" void kernel_launch(void* const* d_in, const int* in_sizes, int n_in,
                              void* d_out, int out_size, void* d_ws, size_t ws_size,
                              hipStream_t stream) {
    const float* q    = (const float*)d_in[0];
    const float* k    = (const float*)d_in[1];
    const float* v    = (const float*)d_in[2];
    const float* g    = (const float*)d_in[3];
    const float* beta = (const float*)d_in[4];
    const float* S0   = (const float*)d_in[5];
    float* out        = (float*)d_out;

    const int BH = in_sizes[3];  // g has B*H elements (2048)
    gdr_cell_kernel<<<BH, 256, 0, stream>>>(q, k, v, g, beta, S0, out, BH);
}